// SpVarModel_13039520711241
// MI455X (gfx1250) — compile-verified
//
#include <hip/hip_runtime.h>

// out[i, 0:8] = params[idx[i]] * xs[i, 0:8]
// N = 4194304 rows, D = 8 floats/row, V = 1048576 params.
//
// Pure HBM-streaming kernel (~288 MB traffic, ~0.03 GFLOP, 0.12 FLOP/B):
// HBM roofline at 23.3 TB/s is ~12 us; no matmul structure so WMMA does not
// apply. CDNA5 memory-path optimizations:
//   - b128 NT loads/stores for the read-once/write-once streams (xs, out)
//   - REGULAR temporal policy on the 4 MB params table -> stays in 192 MB L2
//   - sched_barrier-enforced issue order: idx load -> both xs b128 loads ->
//     (s_wait_loadcnt 2, idx only) -> dependent params gather, so the xs
//     stream is in flight during the index round-trip instead of after it
//   - wave32: 256-thread blocks = 8 waves, 16384 blocks to hide gather latency.

typedef float v4f __attribute__((ext_vector_type(4)));

#define ROW_D 8

#if __has_builtin(__builtin_amdgcn_sched_barrier)
#define SCHED_FENCE() __builtin_amdgcn_sched_barrier(0)
#else
#define SCHED_FENCE() ((void)0)
#endif

__global__ __launch_bounds__(256) void spvar_gather_scale_kernel(
    const float* __restrict__ xs,
    const int*   __restrict__ idx,
    const float* __restrict__ params,
    float*       __restrict__ out,
    int n)
{
    int i = blockIdx.x * blockDim.x + threadIdx.x;
    if (i >= n) return;

    // 1) Index load issued FIRST (oldest outstanding load -> cheapest wait).
    int j = __builtin_nontemporal_load(idx + i);
    SCHED_FENCE();

    // 2) Independent xs stream: two b128 NT loads issued before the gather
    //    stall so they overlap the idx round-trip.
    const v4f* xrow = (const v4f*)(xs + (size_t)i * ROW_D);
    v4f x0 = __builtin_nontemporal_load(xrow + 0);
    v4f x1 = __builtin_nontemporal_load(xrow + 1);
    SCHED_FENCE();

    // 3) Dependent gather: backend can wait with loadcnt<=2 (idx completes
    //    first, loads retire in order). Regular temporal policy keeps the
    //    4 MB table L2-resident.
    float p = params[j];

    v4f r0 = x0 * p;
    v4f r1 = x1 * p;

    // Write-once stream -> two b128 NT stores.
    v4f* orow = (v4f*)(out + (size_t)i * ROW_D);
    __builtin_nontemporal_store(r0, orow + 0);
    __builtin_nontemporal_store(r1, orow + 1);
}

extern "C" void kernel_launch(void* const* d_in, const int* in_sizes, int n_in,
                              void* d_out, int out_size, void* d_ws, size_t ws_size,
                              hipStream_t stream) {
    const float* xs     = (const float*)d_in[0];   // [N, 8] f32
    const int*   idx    = (const int*)d_in[1];     // [N]    int
    const float* params = (const float*)d_in[2];   // [V, 1] f32

    float* out = (float*)d_out;                    // [N, 8] f32

    const int n = in_sizes[1];                     // N rows (idx element count)
    const int block = 256;                         // 8 wave32 waves per block
    const int grid  = (n + block - 1) / block;

    spvar_gather_scale_kernel<<<grid, block, 0, stream>>>(xs, idx, params, out, n);
}